// EnforceDecrease_59150289600719
// MI455X (gfx1250) — compile-verified
//
#include <hip/hip_runtime.h>

#define GLOBAL_AS __attribute__((address_space(1)))
#define LDS_AS    __attribute__((address_space(3)))

typedef int   v4i __attribute__((vector_size(16)));
typedef float v4f __attribute__((vector_size(16)));

// Problem constants from the reference
constexpr int T_SAMP = 121;
constexpr int NCHAN  = 40;
constexpr int NPAR   = 12;
constexpr int TILE   = T_SAMP * NCHAN;     // 4840 floats per spike
constexpr int NVEC   = TILE / 4;           // 1210 float4 chunks (exact)
constexpr int NTHR   = 256;                // 8 wave32s
constexpr int NFULL  = NVEC / NTHR;        // 4 unconditional rounds
constexpr int NTAIL  = NVEC - NFULL * NTHR;// 186 tail lanes
constexpr int NSLICE = 6;                  // 6 T-slices * 40 ch = 240 reduction lanes
constexpr int SLICE_T = (T_SAMP + NSLICE - 1) / NSLICE;  // 21

__global__ __launch_bounds__(NTHR)
void enforce_decrease_kernel(const float* __restrict__ wf,
                             const int*   __restrict__ max_channels,
                             const int*   __restrict__ parents_index,
                             float* __restrict__ out_wf,
                             float* __restrict__ out_ptps)
{
    __shared__ __align__(16) float s_wf[TILE];
    __shared__ float s_pmax[NSLICE * NCHAN];
    __shared__ float s_pmin[NSLICE * NCHAN];
    __shared__ float s_ptp[NCHAN];
    __shared__ float s_scale[NCHAN];

    const int n   = blockIdx.x;
    const int tid = threadIdx.x;

    const float* src = wf + (size_t)n * TILE;

    // ---- Phase 1: stage the 19.4 KB tile into LDS --------------------------
#if __has_builtin(__builtin_amdgcn_global_load_async_to_lds_b128)
    {
        // gfx1250 async data mover: per-lane 16B global->LDS copies (ASYNCcnt).
        // The instruction's immediate byte offset is added to BOTH the global
        // and the LDS address, so one VGPR address pair covers all rounds.
        GLOBAL_AS v4i* g = (GLOBAL_AS v4i*)src + tid;
        LDS_AS    v4i* l = (LDS_AS    v4i*)s_wf + tid;
        __builtin_amdgcn_global_load_async_to_lds_b128(g, l, 0 * NTHR * 16, 0);
        __builtin_amdgcn_global_load_async_to_lds_b128(g, l, 1 * NTHR * 16, 0);
        __builtin_amdgcn_global_load_async_to_lds_b128(g, l, 2 * NTHR * 16, 0);
        __builtin_amdgcn_global_load_async_to_lds_b128(g, l, 3 * NTHR * 16, 0);
        if (tid < NTAIL) {
            __builtin_amdgcn_global_load_async_to_lds_b128(g, l, NFULL * NTHR * 16, 0);
        }
    }
    if (tid == 0) {
        // Overlap: prefetch this spike's parents row while the tile streams in.
        int mc0 = max_channels[n];
        __builtin_prefetch(parents_index + (size_t)mc0 * NCHAN * NPAR, 0, 0);
    }
  #if __has_builtin(__builtin_amdgcn_s_wait_asynccnt)
    __builtin_amdgcn_s_wait_asynccnt(0);
  #else
    asm volatile("s_wait_asynccnt 0" ::: "memory");
  #endif
#else
    // Fallback: plain b128 load + ds store staging.
    for (int j = tid; j < NVEC; j += NTHR) {
        ((v4f*)s_wf)[j] = ((const v4f*)src)[j];
    }
    if (tid == 0) {
        int mc0 = max_channels[n];
        __builtin_prefetch(parents_index + (size_t)mc0 * NCHAN * NPAR, 0, 0);
    }
#endif
    __syncthreads();

    // ---- Phase 2: per-channel max/min over T (ptp) -------------------------
    if (tid < NSLICE * NCHAN) {
        const int ch = tid % NCHAN;
        const int sl = tid / NCHAN;
        int t0 = sl * SLICE_T;
        int t1 = t0 + SLICE_T;
        if (t1 > T_SAMP) t1 = T_SAMP;
        float mx = -__builtin_inff();
        float mn =  __builtin_inff();
        for (int t = t0; t < t1; ++t) {
            float v = s_wf[t * NCHAN + ch];
            mx = fmaxf(mx, v);
            mn = fminf(mn, v);
        }
        s_pmax[sl * NCHAN + ch] = mx;
        s_pmin[sl * NCHAN + ch] = mn;
    }
    __syncthreads();

    if (tid < NCHAN) {
        float mx = s_pmax[tid];
        float mn = s_pmin[tid];
        #pragma unroll
        for (int sl = 1; sl < NSLICE; ++sl) {
            mx = fmaxf(mx, s_pmax[sl * NCHAN + tid]);
            mn = fminf(mn, s_pmin[sl * NCHAN + tid]);
        }
        s_ptp[tid] = mx - mn;
    }
    __syncthreads();

    // ---- Phase 3: parent gather + rescale factor ---------------------------
    if (tid < NCHAN) {
        const int mc = max_channels[n];
        const int* prow = parents_index + ((size_t)mc * NCHAN + tid) * NPAR;
        float pmin = __builtin_inff();  // index == NCHAN maps to the +inf pad
        #pragma unroll
        for (int p = 0; p < NPAR; ++p) {
            int idx = prow[p];
            float v = (idx >= NCHAN) ? __builtin_inff() : s_ptp[idx];
            pmin = fminf(pmin, v);
        }
        const float ptp = s_ptp[tid];
        const float r = fminf(pmin / ptp, 1.0f);
        s_scale[tid] = r;
        out_ptps[(size_t)n * NCHAN + tid] = ptp * r;
    }
    __syncthreads();

    // ---- Phase 4: scale tile and stream out (NT b128 stores) ---------------
    v4f* dst = (v4f*)(out_wf + (size_t)n * TILE);
    #pragma unroll
    for (int k = 0; k < NFULL; ++k) {
        const int j = tid + k * NTHR;
        v4f v = ((const v4f*)s_wf)[j];
        const int ch = (4 * j) % NCHAN;   // NCHAN % 4 == 0: no wrap inside a float4
        v[0] *= s_scale[ch];
        v[1] *= s_scale[ch + 1];
        v[2] *= s_scale[ch + 2];
        v[3] *= s_scale[ch + 3];
        __builtin_nontemporal_store(v, &dst[j]);  // write-once stream: NT hint
    }
    if (tid < NTAIL) {
        const int j = tid + NFULL * NTHR;
        v4f v = ((const v4f*)s_wf)[j];
        const int ch = (4 * j) % NCHAN;
        v[0] *= s_scale[ch];
        v[1] *= s_scale[ch + 1];
        v[2] *= s_scale[ch + 2];
        v[3] *= s_scale[ch + 3];
        __builtin_nontemporal_store(v, &dst[j]);
    }
}

extern "C" void kernel_launch(void* const* d_in, const int* in_sizes, int n_in,
                              void* d_out, int out_size, void* d_ws, size_t ws_size,
                              hipStream_t stream) {
    const float* wf   = (const float*)d_in[0];   // (N, 121, 40) float32
    const int*   mc   = (const int*)d_in[1];     // (N,) integer
    const int*   pidx = (const int*)d_in[2];     // (384, 40, 12) integer
    const int N = in_sizes[1];

    float* out_wf   = (float*)d_out;                       // (N,121,40) flat
    float* out_ptps = out_wf + (size_t)N * T_SAMP * NCHAN; // (N,40) flat

    enforce_decrease_kernel<<<N, NTHR, 0, stream>>>(wf, mc, pidx, out_wf, out_ptps);
}